// GINConcat_83811991814531
// MI455X (gfx1250) — compile-verified
//
#include <hip/hip_runtime.h>

typedef __attribute__((ext_vector_type(2))) float v2f;
typedef __attribute__((ext_vector_type(8))) float v8f;

#define NNODES 100000
#define NEDGES 1600000
#define FIN    64
#define HDIM   128
#define NGR    1024
#define CATDIM 448   // FIN + 3*HDIM
#define LHID   64
#define BNEPS  1e-5f

// ---------------- utility kernels ----------------

__global__ void zero_kernel(float* __restrict__ p, int n) {
  int i = blockIdx.x * blockDim.x + threadIdx.x;
  if (i < n) p[i] = 0.0f;
}

__global__ void copy4_kernel(const float4* __restrict__ s, float4* __restrict__ d, int n4) {
  int i = blockIdx.x * blockDim.x + threadIdx.x;
  if (i < n4) d[i] = s[i];
}

// z[dst[e]][:] += h[src[e]][:]   (one thread = one edge x 4 features)
__global__ void edge_scatter_kernel(const float* __restrict__ h, const int* __restrict__ esrc,
                                    const int* __restrict__ edst, float* __restrict__ z,
                                    int nE, int gshift /* log2(d/4) */, int d) {
  int t = blockIdx.x * blockDim.x + threadIdx.x;
  int total = nE << gshift;
  if (t >= total) return;
  int e = t >> gshift;
  int f = (t & ((1 << gshift) - 1)) << 2;
  int s  = esrc[e];
  int dn = edst[e];
  const float4 v = *(const float4*)(h + (long)s * d + f);
  float* zp = z + (long)dn * d + f;
  atomicAdd(zp + 0, v.x);
  atomicAdd(zp + 1, v.y);
  atomicAdd(zp + 2, v.z);
  atomicAdd(zp + 3, v.w);
}

// pooled[batch[n]][colOff + f] += h[n][f]
__global__ void pool_kernel(const float* __restrict__ h, const int* __restrict__ batch,
                            float* __restrict__ pooled, int rows, int dshift, int colOff) {
  int i = blockIdx.x * blockDim.x + threadIdx.x;
  if (i >= (rows << dshift)) return;
  int n = i >> dshift;
  int f = i & ((1 << dshift) - 1);
  atomicAdd(&pooled[(long)batch[n] * CATDIM + colOff + f], h[i]);
}

// ---------------- WMMA f32 GEMM: Y[M,Hc] = Z[M,K] @ W[Hc,K]^T + bias ----------------
// gridDim.x = M/32 ; blockDim.x = (Hc/16)*32.
// Each wave computes TWO 16x16 M-tiles sharing one B operand per k-step
// (3 b64 loads -> 2 WMMAs), halving redundant weight traffic.
// A lane layout (32-bit 16x4): lanes 0-15 -> K=0,1 ; lanes 16-31 -> K=2,3 (v2f load)
// B lane layout (32-bit 4x16): mirrored -> same addressing with rows<->cols swapped.
__global__ void gemm_bias_wmma(const float* __restrict__ Z, const float* __restrict__ W,
                               const float* __restrict__ bias, float* __restrict__ Y,
                               int K, int Hc) {
  int lane = threadIdx.x & 31;
  int wave = threadIdx.x >> 5;
  int row0 = blockIdx.x << 5;   // 32 rows per block (two 16-row tiles per wave)
  int col0 = wave << 4;
  int mr = lane & 15;           // row-in-tile for A, col-in-tile for B/D
  int kh = (lane >> 4) << 1;    // base K offset of this lane's v2f pair
  const float* zp0 = Z + (long)(row0 + mr) * K + kh;
  const float* zp1 = zp0 + (long)16 * K;
  const float* wp  = W + (long)(col0 + mr) * K + kh;
  v8f acc0 = {};
  v8f acc1 = {};
  for (int k = 0; k < K; k += 4) {
    v2f b  = *(const v2f*)(wp + k);
    v2f a0 = *(const v2f*)(zp0 + k);
    v2f a1 = *(const v2f*)(zp1 + k);
    // 8 args: (neg_a, A, neg_b, B, c_mod, C, reuse_a, reuse_b)
    acc0 = __builtin_amdgcn_wmma_f32_16x16x4_f32(false, a0, false, b, (short)0, acc0,
                                                 false, false);
    acc1 = __builtin_amdgcn_wmma_f32_16x16x4_f32(false, a1, false, b, (short)0, acc1,
                                                 false, false);
  }
  float bv = bias[col0 + mr];
  int rbase = row0 + ((lane >> 4) << 3);   // D: VGPR r -> row rbase + r
  float* yp0 = Y + (long)rbase * Hc + col0 + mr;
  float* yp1 = yp0 + (long)16 * Hc;
#pragma unroll
  for (int r = 0; r < 8; ++r) {
    yp0[(long)r * Hc] = acc0[r] + bv;
    yp1[(long)r * Hc] = acc1[r] + bv;
  }
}

// ---------------- batchnorm (training-mode batch stats) ----------------

__global__ void col_stats_kernel(const float* __restrict__ Y, float* __restrict__ stats,
                                 int rows, int Hc) {
  int col = threadIdx.x;               // blockDim.x == Hc
  float s = 0.f, s2 = 0.f;
  for (int r = blockIdx.x; r < rows; r += gridDim.x) {
    float v = Y[(long)r * Hc + col];
    s += v; s2 += v * v;
  }
  atomicAdd(&stats[col], s);
  atomicAdd(&stats[Hc + col], s2);
}

__global__ void bn_relu_kernel(float* __restrict__ Y, const float* __restrict__ stats,
                               const float* __restrict__ gamma, const float* __restrict__ beta,
                               int total, int Hc, float invRows) {
  int i = blockIdx.x * blockDim.x + threadIdx.x;
  if (i >= total) return;
  int col = i & (Hc - 1);
  float m  = stats[col] * invRows;
  float v  = stats[Hc + col] * invRows - m * m;   // biased variance
  float sc = gamma[col] * rsqrtf(v + BNEPS);
  float val = (Y[i] - m) * sc + beta[col];
  Y[i] = fmaxf(val, 0.0f);
}

// ---------------- final tiny linear: out[g] = relu(h1[g]) . w + b ----------------
__global__ void final_kernel(const float* __restrict__ h1, const float* __restrict__ w,
                             const float* __restrict__ b, float* __restrict__ out) {
  int g = blockIdx.x * blockDim.x + threadIdx.x;
  if (g >= NGR) return;
  float acc = b[0];
  const float* hp = h1 + g * LHID;
#pragma unroll 8
  for (int k = 0; k < LHID; ++k) acc += fmaxf(hp[k], 0.0f) * w[k];
  out[g] = acc;
}

// ---------------- driver ----------------

static inline int cdiv_l(long a, long b) { return (int)((a + b - 1) / b); }

extern "C" void kernel_launch(void* const* d_in, const int* in_sizes, int n_in,
                              void* d_out, int out_size, void* d_ws, size_t ws_size,
                              hipStream_t stream) {
  // inputs: x, edge_index, batch, then params in insertion order:
  //   per conv: w0,b0,w1,b1,mlp_bn_g,mlp_bn_b,bn_g,bn_b ; then lw0,lb0,lw1,lb1
  const float* x   = (const float*)d_in[0];
  const int* ei    = (const int*)d_in[1];
  const int* esrc  = ei;
  const int* edst  = ei + NEDGES;
  const int* batch = (const int*)d_in[2];
  const float *cw0[3], *cb0[3], *cw1[3], *cb1[3], *cmg[3], *cmb[3], *cbg[3], *cbb[3];
  for (int i = 0; i < 3; ++i) {
    int base = 3 + i * 8;
    cw0[i] = (const float*)d_in[base + 0];
    cb0[i] = (const float*)d_in[base + 1];
    cw1[i] = (const float*)d_in[base + 2];
    cb1[i] = (const float*)d_in[base + 3];
    cmg[i] = (const float*)d_in[base + 4];
    cmb[i] = (const float*)d_in[base + 5];
    cbg[i] = (const float*)d_in[base + 6];
    cbb[i] = (const float*)d_in[base + 7];
  }
  const float* lw0 = (const float*)d_in[27];
  const float* lb0 = (const float*)d_in[28];
  const float* lw1 = (const float*)d_in[29];
  const float* lb1 = (const float*)d_in[30];

  float* ws = (float*)d_ws;
  const size_t SZ = (size_t)NNODES * HDIM;
  float* zbuf   = ws;              // [N,128] aggregated features
  float* ybuf   = ws + SZ;         // [N,128] hidden after lin0
  float* hbuf   = ws + 2 * SZ;     // [N,128] conv output (next h)
  float* pooled = ws + 3 * SZ;     // [G,448]
  float* stats  = pooled + (size_t)NGR * CATDIM;  // 256 floats
  float* h1     = stats + 2 * HDIM;               // [G,64]

  // readout slot 0: pooled raw features
  zero_kernel<<<cdiv_l((long)NGR * CATDIM, 256), 256, 0, stream>>>(pooled, NGR * CATDIM);
  pool_kernel<<<cdiv_l((long)NNODES * FIN, 256), 256, 0, stream>>>(x, batch, pooled, NNODES, 6, 0);

  const float* hcur = x;
  for (int i = 0; i < 3; ++i) {
    int d    = (i == 0) ? FIN : HDIM;
    int dlog = (i == 0) ? 6 : 7;
    long nd  = (long)NNODES * d;

    // z = h + segment_sum(h[src], dst)
    copy4_kernel<<<cdiv_l(nd / 4, 256), 256, 0, stream>>>((const float4*)hcur, (float4*)zbuf,
                                                          (int)(nd / 4));
    edge_scatter_kernel<<<cdiv_l((long)NEDGES << (dlog - 2), 256), 256, 0, stream>>>(
        hcur, esrc, edst, zbuf, NEDGES, dlog - 2, d);

    // y = z @ w0^T + b0 ; BN(mlp) + relu
    gemm_bias_wmma<<<NNODES / 32, (HDIM / 16) * 32, 0, stream>>>(zbuf, cw0[i], cb0[i], ybuf, d, HDIM);
    zero_kernel<<<1, 256, 0, stream>>>(stats, 2 * HDIM);
    col_stats_kernel<<<512, HDIM, 0, stream>>>(ybuf, stats, NNODES, HDIM);
    bn_relu_kernel<<<cdiv_l((long)NNODES * HDIM, 256), 256, 0, stream>>>(
        ybuf, stats, cmg[i], cmb[i], NNODES * HDIM, HDIM, 1.0f / NNODES);

    // h = y @ w1^T + b1 ; outer BN + relu
    gemm_bias_wmma<<<NNODES / 32, (HDIM / 16) * 32, 0, stream>>>(ybuf, cw1[i], cb1[i], hbuf, HDIM, HDIM);
    zero_kernel<<<1, 256, 0, stream>>>(stats, 2 * HDIM);
    col_stats_kernel<<<512, HDIM, 0, stream>>>(hbuf, stats, NNODES, HDIM);
    bn_relu_kernel<<<cdiv_l((long)NNODES * HDIM, 256), 256, 0, stream>>>(
        hbuf, stats, cbg[i], cbb[i], NNODES * HDIM, HDIM, 1.0f / NNODES);

    // readout slot for this conv
    pool_kernel<<<cdiv_l((long)NNODES * HDIM, 256), 256, 0, stream>>>(
        hbuf, batch, pooled, NNODES, 7, FIN + i * HDIM);

    hcur = hbuf;
  }

  // h1 = pooled @ lw0^T + lb0  ([1024,448] x [64,448]^T), same WMMA kernel
  gemm_bias_wmma<<<NGR / 32, (LHID / 16) * 32, 0, stream>>>(pooled, lw0, lb0, h1, CATDIM, LHID);
  // out = relu(h1) @ lw1^T + lb1
  final_kernel<<<cdiv_l(NGR, 256), 256, 0, stream>>>(h1, lw1, lb1, (float*)d_out);
}